// LSTM_forward_model_50689204027636
// MI455X (gfx1250) — compile-verified
//
#include <hip/hip_runtime.h>
#include <math.h>

// Problem dims (match reference)
#define B_  256
#define T_  200
#define I_  32
#define H_  512
#define O_  7
#define MT  2      // batch tiles (16 rows each) per wave

typedef __attribute__((ext_vector_type(2))) float v2f;
typedef __attribute__((ext_vector_type(8))) float v8f;

__device__ __forceinline__ float sigmoidf_(float x) { return 1.0f / (1.0f + expf(-x)); }

#define WMMA_F32(A, Bv, C) \
    __builtin_amdgcn_wmma_f32_16x16x4_f32(false, (A), false, (Bv), (short)0, (C), false, false)

// One fused LSTM timestep for one layer:
//   gates = [x_t | h_prev] @ [w_ih | w_hh]^T + (b_ih + b_hh)
//   c = sigmoid(f)*c + sigmoid(i)*tanh(g);  h = sigmoid(o)*tanh(c)
// Grid: (H/16, B/(16*MT)), block = one wave32 (EXEC all ones at every WMMA).
// Each wave owns MT 16x16 (batch x hidden) tiles for all 4 gates.
// V_WMMA_F32_16X16X4_F32 operand layouts (ISA 7.12.2):
//   A 16x4 : lane m=lane&15, half kg=lane>>4; VGPR v holds A[m][v+2*kg]
//   B 4x16 : lane n=lane&15; VGPR v holds B[v+2*kg][n]
//   C 16x16: VGPR v holds C[v+8*kg][n]
// K-loop is software-pipelined (next chunk's 6 loads issued before current
// chunk's 8 WMMAs) so waits land on loads issued a full iteration earlier.
__global__ __launch_bounds__(32)
void lstm_step_wmma(const float* __restrict__ xin, int xstride, int Din,
                    const float* __restrict__ hprev,
                    const float* __restrict__ w_ih,   // [4H, Din] row-major
                    const float* __restrict__ w_hh,   // [4H, H]   row-major
                    const float* __restrict__ b_ih,
                    const float* __restrict__ b_hh,
                    float* __restrict__ c,            // [B, H] in/out
                    float* __restrict__ hout,         // [B, H]
                    int first_step)
{
    const int lane = threadIdx.x;         // 0..31
    const int m    = lane & 15;
    const int kg   = lane >> 4;           // lane half selects K pair / M half
    const int nb   = blockIdx.x * 16;     // hidden-dim tile base
    const int mb   = blockIdx.y * (16 * MT); // batch tile base
    const int col  = nb + m;              // this lane's N column

    // Accumulators: [gate][mtile], initialized with combined bias
    v8f acc[4][MT];
#pragma unroll
    for (int g = 0; g < 4; ++g) {
        const float bb = b_ih[g * H_ + col] + b_hh[g * H_ + col];
#pragma unroll
        for (int mt = 0; mt < MT; ++mt)
#pragma unroll
            for (int v = 0; v < 8; ++v) acc[g][mt][v] = bb;
    }

    // ---------------- input projection: K over Din ----------------
    {
        int aoff[MT], boff[4];
#pragma unroll
        for (int mt = 0; mt < MT; ++mt) aoff[mt] = (mb + mt * 16 + m) * xstride + 2 * kg;
#pragma unroll
        for (int g = 0; g < 4; ++g)     boff[g]  = (g * H_ + col) * Din + 2 * kg;

        v2f aC[MT], bC[4];
#pragma unroll
        for (int mt = 0; mt < MT; ++mt) aC[mt] = *(const v2f*)(xin + aoff[mt]);
#pragma unroll
        for (int g = 0; g < 4; ++g)     bC[g]  = *(const v2f*)(w_ih + boff[g]);

        for (int k0 = 4; k0 < Din; k0 += 4) {
            v2f aN[MT], bN[4];
#pragma unroll
            for (int mt = 0; mt < MT; ++mt) aN[mt] = *(const v2f*)(xin + aoff[mt] + k0);
#pragma unroll
            for (int g = 0; g < 4; ++g)     bN[g]  = *(const v2f*)(w_ih + boff[g] + k0);
#pragma unroll
            for (int g = 0; g < 4; ++g)
#pragma unroll
                for (int mt = 0; mt < MT; ++mt) acc[g][mt] = WMMA_F32(aC[mt], bC[g], acc[g][mt]);
#pragma unroll
            for (int mt = 0; mt < MT; ++mt) aC[mt] = aN[mt];
#pragma unroll
            for (int g = 0; g < 4; ++g)     bC[g]  = bN[g];
        }
#pragma unroll
        for (int g = 0; g < 4; ++g)
#pragma unroll
            for (int mt = 0; mt < MT; ++mt) acc[g][mt] = WMMA_F32(aC[mt], bC[g], acc[g][mt]);
    }

    // ---------------- recurrent projection: K over H ----------------
    if (!first_step) {
        int aoff[MT], boff[4];
#pragma unroll
        for (int mt = 0; mt < MT; ++mt) aoff[mt] = (mb + mt * 16 + m) * H_ + 2 * kg;
#pragma unroll
        for (int g = 0; g < 4; ++g)     boff[g]  = (g * H_ + col) * H_ + 2 * kg;

        v2f aC[MT], bC[4];
#pragma unroll
        for (int mt = 0; mt < MT; ++mt) aC[mt] = *(const v2f*)(hprev + aoff[mt]);
#pragma unroll
        for (int g = 0; g < 4; ++g)     bC[g]  = *(const v2f*)(w_hh + boff[g]);

        for (int k0 = 4; k0 < H_; k0 += 4) {
            v2f aN[MT], bN[4];
#pragma unroll
            for (int mt = 0; mt < MT; ++mt) aN[mt] = *(const v2f*)(hprev + aoff[mt] + k0);
#pragma unroll
            for (int g = 0; g < 4; ++g)     bN[g]  = *(const v2f*)(w_hh + boff[g] + k0);
#pragma unroll
            for (int g = 0; g < 4; ++g)
#pragma unroll
                for (int mt = 0; mt < MT; ++mt) acc[g][mt] = WMMA_F32(aC[mt], bC[g], acc[g][mt]);
#pragma unroll
            for (int mt = 0; mt < MT; ++mt) aC[mt] = aN[mt];
#pragma unroll
            for (int g = 0; g < 4; ++g)     bC[g]  = bN[g];
        }
#pragma unroll
        for (int g = 0; g < 4; ++g)
#pragma unroll
            for (int mt = 0; mt < MT; ++mt) acc[g][mt] = WMMA_F32(aC[mt], bC[g], acc[g][mt]);
    }

    // ---------------- epilogue: gate activations + state update ----------------
#pragma unroll
    for (int mt = 0; mt < MT; ++mt) {
#pragma unroll
        for (int v = 0; v < 8; ++v) {
            const int row = mb + mt * 16 + v + 8 * kg;
            const int idx = row * H_ + col;
            const float iv = sigmoidf_(acc[0][mt][v]);
            const float fv = sigmoidf_(acc[1][mt][v]);
            const float gv = tanhf(acc[2][mt][v]);
            const float ov = sigmoidf_(acc[3][mt][v]);
            const float cold = first_step ? 0.0f : c[idx];
            const float cn = fv * cold + iv * gv;
            c[idx]    = cn;
            hout[idx] = ov * tanhf(cn);
        }
    }
}

// FC head: y = h_last @ w_fc^T + b_fc; p = sigmoid(y[:,:3]); r = normalize(tanh(y[:,3:]))
__global__ void fc_head_kernel(const float* __restrict__ hlast,
                               const float* __restrict__ w_fc,
                               const float* __restrict__ b_fc,
                               float* __restrict__ out)
{
    const int b = blockIdx.x * blockDim.x + threadIdx.x;
    if (b >= B_) return;
    const float* h = hlast + b * H_;
    float y[O_];
#pragma unroll
    for (int o = 0; o < O_; ++o) {
        float s = b_fc[o];
        const float* w = w_fc + o * H_;
        for (int k = 0; k < H_; ++k) s += h[k] * w[k];
        y[o] = s;
    }
    out[b * 3 + 0] = sigmoidf_(y[0]);
    out[b * 3 + 1] = sigmoidf_(y[1]);
    out[b * 3 + 2] = sigmoidf_(y[2]);
    const float r0 = tanhf(y[3]);
    const float r1 = tanhf(y[4]);
    const float r2 = tanhf(y[5]);
    const float r3 = tanhf(y[6]);
    const float nrm = sqrtf(r0 * r0 + r1 * r1 + r2 * r2 + r3 * r3);
    const float inv = 1.0f / fmaxf(nrm, 1e-12f);
    float* r = out + B_ * 3 + b * 4;
    r[0] = r0 * inv;
    r[1] = r1 * inv;
    r[2] = r2 * inv;
    r[3] = r3 * inv;
}

extern "C" void kernel_launch(void* const* d_in, const int* in_sizes, int n_in,
                              void* d_out, int out_size, void* d_ws, size_t ws_size,
                              hipStream_t stream)
{
    const float* x     = (const float*)d_in[0];   // [B, T, I]
    const float* w_ih0 = (const float*)d_in[1];   // [4H, I]
    const float* w_hh0 = (const float*)d_in[2];   // [4H, H]
    const float* b_ih0 = (const float*)d_in[3];
    const float* b_hh0 = (const float*)d_in[4];
    const float* w_ih1 = (const float*)d_in[5];   // [4H, H]
    const float* w_hh1 = (const float*)d_in[6];   // [4H, H]
    const float* b_ih1 = (const float*)d_in[7];
    const float* b_hh1 = (const float*)d_in[8];
    const float* w_fc  = (const float*)d_in[9];   // [O, H]
    const float* b_fc  = (const float*)d_in[10];

    float* ws = (float*)d_ws;
    const size_t BH = (size_t)B_ * H_;            // 131072 floats
    float* h0[2] = { ws + 0 * BH, ws + 1 * BH };  // layer-0 hidden ping-pong
    float* h1[2] = { ws + 2 * BH, ws + 3 * BH };  // layer-1 hidden ping-pong
    float* c0    =   ws + 4 * BH;                 // layer-0 cell state
    float* c1    =   ws + 5 * BH;                 // layer-1 cell state
    // Scratch: 3 MiB total. No zero-init needed: first_step flags make t==0
    // ignore h_prev/c contents, so every graph replay recomputes from scratch.

    const dim3 grid(H_ / 16, B_ / (16 * MT));     // 32 x 8 tiles
    const dim3 blk(32);                           // one wave32 per block

    for (int t = 0; t < T_; ++t) {
        const int cur = t & 1, prev = cur ^ 1;
        lstm_step_wmma<<<grid, blk, 0, stream>>>(
            x + (size_t)t * I_, T_ * I_, I_,
            h0[prev], w_ih0, w_hh0, b_ih0, b_hh0, c0, h0[cur], t == 0);
        lstm_step_wmma<<<grid, blk, 0, stream>>>(
            h0[cur], H_, H_,
            h1[prev], w_ih1, w_hh1, b_ih1, b_hh1, c1, h1[cur], t == 0);
    }

    fc_head_kernel<<<(B_ + 63) / 64, 64, 0, stream>>>(
        h1[(T_ - 1) & 1], w_fc, b_fc, (float*)d_out);
}